// GraphConvolution_86268713107474
// MI455X (gfx1250) — compile-verified
//
#include <hip/hip_runtime.h>

typedef __attribute__((ext_vector_type(2))) float v2f;
typedef __attribute__((ext_vector_type(8))) float v8f;

#define D_IN  128
#define D_OUT 64

// ---------------------------------------------------------------------------
// Kernel 1: hidden = x @ W + b   via V_WMMA_F32_16X16X4_F32 (full fp32 path)
// One wave computes a 16(M) x 64(N) tile: 4 accumulators, K loop of 32 steps.
// ---------------------------------------------------------------------------
__global__ __launch_bounds__(128)
void gc_gemm_bias_wmma(const float* __restrict__ x,
                       const float* __restrict__ W,
                       const float* __restrict__ b,
                       float* __restrict__ hidden,
                       int n_nodes) {
    const int lane  = threadIdx.x & 31;
    const int wave  = blockIdx.x * (blockDim.x >> 5) + (threadIdx.x >> 5);
    const int m0    = wave * 16;
    if (m0 >= n_nodes) return;           // wave-uniform exit -> EXEC all-1s for WMMA

    const int half = lane >> 4;          // 0: K even pair, 1: K odd pair (per ISA 16x4 layout)
    const int l15  = lane & 15;          // M index for A, N index for B/C/D

    // Accumulators pre-loaded with bias (bias depends only on N = l15)
    v8f acc[4];
#pragma unroll
    for (int nt = 0; nt < 4; ++nt) {
        const float bv = b[nt * 16 + l15];
#pragma unroll
        for (int i = 0; i < 8; ++i) acc[nt][i] = bv;
    }

    const float* xrow = x + (size_t)(m0 + l15) * D_IN + 2 * half;

#pragma unroll 4
    for (int k = 0; k < D_IN; k += 4) {
        // A fragment 16x4: lane l -> x[m0 + (l&15)][k + 2*(l>>4) + {0,1}]
        const v2f a = *(const v2f*)(xrow + k);
        const float* wk = W + (size_t)(k + 2 * half) * D_OUT + l15;
#pragma unroll
        for (int nt = 0; nt < 4; ++nt) {
            // B fragment 4x16: lane l -> W[k + 2*(l>>4) + {0,1}][n0 + (l&15)]
            v2f bf;
            bf.x = wk[nt * 16];
            bf.y = wk[nt * 16 + D_OUT];
            acc[nt] = __builtin_amdgcn_wmma_f32_16x16x4_f32(
                /*neg_a=*/false, a, /*neg_b=*/false, bf,
                /*c_mod=*/(short)0, acc[nt],
                /*reuse_a=*/false, /*reuse_b=*/false);
        }
    }

    // C/D layout: VGPR i, lanes0-15 -> M = m0+i, lanes16-31 -> M = m0+i+8, N = l15
#pragma unroll
    for (int nt = 0; nt < 4; ++nt) {
#pragma unroll
        for (int i = 0; i < 8; ++i) {
            const int m = m0 + i + 8 * half;
            hidden[(size_t)m * D_OUT + nt * 16 + l15] = acc[nt][i];
        }
    }
}

// ---------------------------------------------------------------------------
// Kernel 2: zero the aggregation buffer (harness poisons d_out with 0xAA)
// ---------------------------------------------------------------------------
__global__ void gc_zero(float* __restrict__ out, int n) {
    int i = blockIdx.x * blockDim.x + threadIdx.x;
    if (i < n) out[i] = 0.0f;
}

// ---------------------------------------------------------------------------
// Kernel 3: edge scatter. One thread per (edge, channel).
// out[row[e]][n] += w[e] * hidden[col[e]][n]   (L2 atomic adds; table is
// L2-resident: 12.8 MB gather table, 12.8 MB scatter target vs 192 MB L2)
// ---------------------------------------------------------------------------
__global__ __launch_bounds__(256)
void gc_scatter_edges(const int* __restrict__ erow,
                      const int* __restrict__ ecol,
                      const float* __restrict__ ew,
                      const float* __restrict__ hidden,
                      float* __restrict__ out,
                      int n_edges) {
    const long long gid = (long long)blockIdx.x * blockDim.x + threadIdx.x;
    const int e = (int)(gid >> 6);        // D_OUT == 64 channels per edge
    if (e >= n_edges) return;
    const int n = (int)(gid & 63);

    const int   r = erow[e];
    const int   c = ecol[e];
    const float w = ew[e];
    const float v = w * hidden[(size_t)c * D_OUT + n];
    atomicAdd(out + (size_t)r * D_OUT + n, v);
}

// ---------------------------------------------------------------------------
// Kernel 4: ReLU in place (runs after all scatter atomics have drained)
// ---------------------------------------------------------------------------
__global__ void gc_relu(float* __restrict__ out, int n) {
    int i = blockIdx.x * blockDim.x + threadIdx.x;
    if (i < n) out[i] = fmaxf(out[i], 0.0f);
}

// ---------------------------------------------------------------------------
// Launcher. Inputs: x, edge_row, edge_col, edge_weight, W, b
// ---------------------------------------------------------------------------
extern "C" void kernel_launch(void* const* d_in, const int* in_sizes, int n_in,
                              void* d_out, int out_size, void* d_ws, size_t ws_size,
                              hipStream_t stream) {
    const float* x    = (const float*)d_in[0];
    const int*   erow = (const int*)  d_in[1];
    const int*   ecol = (const int*)  d_in[2];
    const float* ew   = (const float*)d_in[3];
    const float* W    = (const float*)d_in[4];
    const float* b    = (const float*)d_in[5];
    float*       out  = (float*)d_out;

    const int n_nodes = in_sizes[0] / D_IN;   // 50000
    const int n_edges = in_sizes[1];          // 800000

    float* hidden = (float*)d_ws;             // n_nodes * D_OUT floats = 12.8 MB

    // 1) GEMM + bias: one wave per 16-row tile; 4 waves (128 threads) per block
    {
        const int waves  = (n_nodes + 15) / 16;           // 3125
        const int blocks = (waves + 3) / 4;               // 782
        gc_gemm_bias_wmma<<<blocks, 128, 0, stream>>>(x, W, b, hidden, n_nodes);
    }

    // 2) Zero output
    {
        const int n = n_nodes * D_OUT;
        gc_zero<<<(n + 255) / 256, 256, 0, stream>>>(out, n);
    }

    // 3) Scatter-add over edges
    {
        const long long total = (long long)n_edges * D_OUT;
        const int blocks = (int)((total + 255) / 256);    // 200000
        gc_scatter_edges<<<blocks, 256, 0, stream>>>(erow, ecol, ew, hidden, out, n_edges);
    }

    // 4) ReLU
    {
        const int n = n_nodes * D_OUT;
        gc_relu<<<(n + 255) / 256, 256, 0, stream>>>(out, n);
    }
}